// NTXentLossChildrenClean_85624468013604
// MI455X (gfx1250) — compile-verified
//
#include <hip/hip_runtime.h>

// NT-Xent children loss, restructured as two 1024x1024x512 FP32 WMMA GEMMs
// (S_j = ci*cj^T, S_k = ci*ci^T), per-child squared norms, then a tiny
// gather + masked logsumexp pass. Heavy math via V_WMMA_F32_16X16X4_F32.
// This revision: 32x32 macro-tile per wave with 4 accumulators -> each
// operand load feeds two WMMAs (1 wmma per b64 load instead of 0.5),
// halving L2 traffic for the compute-dominant GEMM stage.

#define BB 64      // batch
#define CC 16      // children per sample
#define DD 512     // feature dim
#define RR 1024    // BB*CC rows of the flattened child matrices
#define KK DD
#define INV_TEMP 10.0f     // 1 / 0.1
#define EPSI 1e-8f
#define NEGINF (-1e30f)

typedef float v2f __attribute__((ext_vector_type(2)));
typedef float v8f __attribute__((ext_vector_type(8)));

// ---------------------------------------------------------------------------
// Kernel A: squared norms of each 512-float child row (2048 rows total).
// One wave per row; wave32 shuffle reduction.
// ---------------------------------------------------------------------------
__global__ void ntx_sqnorm_kernel(const float* __restrict__ ci,
                                  const float* __restrict__ cj,
                                  float* __restrict__ sqnI,
                                  float* __restrict__ sqnJ) {
    const int r    = blockIdx.x;          // 0..2047
    const int lane = threadIdx.x;         // 0..31
    const float* row = (r < RR) ? (ci + (size_t)r * KK)
                                : (cj + (size_t)(r - RR) * KK);
    float s = 0.0f;
    #pragma unroll 4
    for (int k = lane; k < KK; k += 32) {
        float x = row[k];
        s += x * x;
    }
    #pragma unroll
    for (int off = 16; off > 0; off >>= 1)
        s += __shfl_down(s, off, 32);
    if (lane == 0) {
        if (r < RR) sqnI[r] = s; else sqnJ[r - RR] = s;
    }
}

// ---------------------------------------------------------------------------
// Kernel B: S[z] = ci_flat(1024x512) * X^T, X = cj_flat (z=0) or ci_flat (z=1).
// One wave computes a 32x32 output macro-tile (4 x 16x16 WMMA accumulators).
// f32 16x16x4 operand layouts (wave32):
//   A: lane L holds M=L%16, k = vgpr + 2*(L>=16)  -> float2 @ k0 + 2*hi
//   B: lane L holds N=L%16, k = vgpr + 2*(L>=16)  -> identical addressing
//   C/D: lane L holds N=L%16, m = vgpr + 8*(L>=16)
// Each (a0,a1,b0,b1) load quartet feeds 4 independent WMMAs (no D->A/B RAW).
// ---------------------------------------------------------------------------
__global__ void ntx_wmma_gemm_kernel(const float* __restrict__ ci,
                                     const float* __restrict__ cj,
                                     float* __restrict__ S) {
    const int lane = threadIdx.x;         // 32 threads = 1 wave
    const int l16  = lane & 15;
    const int hi   = lane >> 4;           // 0 or 1 (half-wave)
    const int nT   = blockIdx.x;          // N macro-tile (32 cols)
    const int mT   = blockIdx.y;          // M macro-tile (32 rows)
    const int z    = blockIdx.z;          // 0 -> S_j (B=cj), 1 -> S_k (B=ci)

    const float* Bmat = z ? ci : cj;
    float* out = S + (size_t)z * RR * RR;

    const float* arow0 = ci   + (size_t)(mT * 32 + l16) * KK + 2 * hi;
    const float* arow1 = arow0 + (size_t)16 * KK;
    const float* brow0 = Bmat + (size_t)(nT * 32 + l16) * KK + 2 * hi;
    const float* brow1 = brow0 + (size_t)16 * KK;

    v8f acc00 = {}, acc01 = {}, acc10 = {}, acc11 = {};
    for (int k0 = 0; k0 < KK; k0 += 32) {
        // prefetch the next 32-float K-chunk of all four operand rows
        if (k0 + 32 < KK) {
            __builtin_prefetch(arow0 + k0 + 32, 0, 3);
            __builtin_prefetch(arow1 + k0 + 32, 0, 3);
            __builtin_prefetch(brow0 + k0 + 32, 0, 3);
            __builtin_prefetch(brow1 + k0 + 32, 0, 3);
        }
        #pragma unroll
        for (int kk = 0; kk < 32; kk += 4) {
            v2f a0 = *(const v2f*)(arow0 + k0 + kk);
            v2f a1 = *(const v2f*)(arow1 + k0 + kk);
            v2f b0 = *(const v2f*)(brow0 + k0 + kk);
            v2f b1 = *(const v2f*)(brow1 + k0 + kk);
            // 8 args: (neg_a, A, neg_b, B, c_mod, C, reuse_a, reuse_b)
            acc00 = __builtin_amdgcn_wmma_f32_16x16x4_f32(
                false, a0, false, b0, (short)0, acc00, false, false);
            acc01 = __builtin_amdgcn_wmma_f32_16x16x4_f32(
                false, a0, false, b1, (short)0, acc01, false, false);
            acc10 = __builtin_amdgcn_wmma_f32_16x16x4_f32(
                false, a1, false, b0, (short)0, acc10, false, false);
            acc11 = __builtin_amdgcn_wmma_f32_16x16x4_f32(
                false, a1, false, b1, (short)0, acc11, false, false);
        }
    }

    const size_t m0   = (size_t)(mT * 32 + 8 * hi);       // M rows of top subtiles
    const size_t m1   = m0 + 16;                          // M rows of bottom subtiles
    const size_t n0   = (size_t)(nT * 32 + l16);          // N col of left subtiles
    const size_t n1   = n0 + 16;                          // N col of right subtiles
    #pragma unroll
    for (int v = 0; v < 8; ++v) {
        out[(m0 + (size_t)v) * RR + n0] = acc00[v];
        out[(m0 + (size_t)v) * RR + n1] = acc01[v];
        out[(m1 + (size_t)v) * RR + n0] = acc10[v];
        out[(m1 + (size_t)v) * RR + n1] = acc11[v];
    }
}

// ---------------------------------------------------------------------------
// Kernel C: per-anchor logits + masked logsumexp.
// Block i (128 threads): t<64 -> sim_j column t, t>=64 -> sim_k column t-64.
// ---------------------------------------------------------------------------
__global__ void ntx_logits_kernel(const float* __restrict__ S,
                                  const float* __restrict__ sqnI,
                                  const float* __restrict__ sqnJ,
                                  const int* __restrict__ pids,
                                  const int* __restrict__ njIdx,
                                  const int* __restrict__ nkIdx,
                                  float* __restrict__ lossPart) {
    const int i = blockIdx.x;             // anchor
    const int t = threadIdx.x;            // 0..127
    __shared__ float lg[1 + 2 * BB];

    // anchor norm (cheap, computed redundantly per thread)
    float sqi = 0.0f;
    #pragma unroll
    for (int c = 0; c < CC; ++c) sqi += sqnI[i * CC + c];
    const float normI = fmaxf(sqrtf(sqi), EPSI);

    const int  j   = t & 63;
    const bool isK = (t >= 64);
    const float* Sm  = S + (isK ? (size_t)RR * RR : 0);
    const float* sqv = isK ? sqnI : sqnJ;     // gathered rows come from ci (k) / cj (j)
    const int*   idx = (isK ? nkIdx : njIdx) + ((size_t)i * BB + j) * CC;

    float num = 0.0f, sqs = 0.0f;
    #pragma unroll
    for (int c = 0; c < CC; ++c) {
        const int nc = idx[c];
        num += Sm[(size_t)(i * CC + c) * RR + (size_t)(j * CC + nc)];
        sqs += sqv[j * CC + nc];
    }
    const float sim   = num / (normI * fmaxf(sqrtf(sqs), EPSI));
    const bool  valid = (j != i) && (pids[j] != pids[i]);
    lg[1 + t] = (valid ? sim : NEGINF) * INV_TEMP;

    if (t == 0) {
        float numP = 0.0f, sqj = 0.0f;
        #pragma unroll
        for (int c = 0; c < CC; ++c) {
            numP += S[(size_t)(i * CC + c) * RR + (size_t)(i * CC + c)];
            sqj  += sqnJ[i * CC + c];
        }
        const float normJ = fmaxf(sqrtf(sqj), EPSI);
        lg[0] = (numP / (normI * normJ)) * INV_TEMP;
    }
    __syncthreads();

    if (t == 0) {
        float m = lg[0];
        for (int x = 1; x < 1 + 2 * BB; ++x) m = fmaxf(m, lg[x]);
        float sum = 0.0f;
        for (int x = 0; x < 1 + 2 * BB; ++x) sum += expf(lg[x] - m);
        lossPart[i] = -lg[0] + m + logf(sum);
    }
}

// ---------------------------------------------------------------------------
// Kernel D: deterministic final reduction -> d_out[0] = sum / (2*B)
// ---------------------------------------------------------------------------
__global__ void ntx_finalize_kernel(const float* __restrict__ lossPart,
                                    float* __restrict__ out) {
    if (threadIdx.x == 0) {
        float s = 0.0f;
        for (int i = 0; i < BB; ++i) s += lossPart[i];
        out[0] = s / (2.0f * (float)BB);
    }
}

// ---------------------------------------------------------------------------
extern "C" void kernel_launch(void* const* d_in, const int* in_sizes, int n_in,
                              void* d_out, int out_size, void* d_ws, size_t ws_size,
                              hipStream_t stream) {
    (void)in_sizes; (void)n_in; (void)out_size; (void)ws_size;
    const float* ci   = (const float*)d_in[0];   // children_is [64,16,512] f32
    const float* cj   = (const float*)d_in[1];   // children_js [64,16,512] f32
    const int*   pids = (const int*)d_in[2];     // partnet_ids [64] i32
    const int*   nj   = (const int*)d_in[3];     // neg_idx_j [64,64,16] i32
    const int*   nk   = (const int*)d_in[4];     // neg_idx_k [64,64,16] i32
    float* out = (float*)d_out;

    // workspace: S (2 x 1024 x 1024 f32 = 8 MB) | sqnI[1024] | sqnJ[1024] | lossPart[64]
    float* S        = (float*)d_ws;
    float* sqnI     = S + 2 * (size_t)RR * RR;
    float* sqnJ     = sqnI + RR;
    float* lossPart = sqnJ + RR;

    ntx_sqnorm_kernel   <<<dim3(2 * RR),            dim3(32),  0, stream>>>(ci, cj, sqnI, sqnJ);
    ntx_wmma_gemm_kernel<<<dim3(RR/32, RR/32, 2),   dim3(32),  0, stream>>>(ci, cj, S);
    ntx_logits_kernel   <<<dim3(BB),                dim3(128), 0, stream>>>(S, sqnI, sqnJ, pids, nj, nk, lossPart);
    ntx_finalize_kernel <<<dim3(1),                 dim3(32),  0, stream>>>(lossPart, out);
}